// Samples_Loss_57037165691500
// MI455X (gfx1250) — compile-verified
//
#include <hip/hip_runtime.h>
#include <cmath>
#include <vector>

// Sinkhorn (geomloss SamplesLoss, p=2, blur=0.01, scaling=0.9) on MI455X.
// B=16, N=M=1024, D=2.  Core softmin tile = V_WMMA_F32_16X16X4_F32:
//   D(16m x 16n) = A(16m x 4k = y/eps) * B(4k x 16n = x) + C(h'_m broadcast)
// followed by an online logsumexp over m (8 regs/lane + lane-half merge).

typedef float v2f __attribute__((ext_vector_type(2)));
typedef float v8f __attribute__((ext_vector_type(8)));

#define BATCH 16
#define NPTS  1024   // N == M

// ---------------------------------------------------------------- prep
__global__ __launch_bounds__(256) void sk_prep(
    const float* __restrict__ a, const float* __restrict__ x,
    const float* __restrict__ b, const float* __restrict__ y,
    float* __restrict__ sx, float* __restrict__ sy,
    float* __restrict__ alog, float* __restrict__ blog)
{
    int i = blockIdx.x * blockDim.x + threadIdx.x;   // 0 .. B*N-1
    if (i < BATCH * NPTS) {
        float x0 = x[2 * i], x1 = x[2 * i + 1];
        sx[i]   = x0 * x0 + x1 * x1;                 // |x|^2 (cost uses 0.5*)
        alog[i] = __logf(a[i]);
        float y0 = y[2 * i], y1 = y[2 * i + 1];
        sy[i]   = y0 * y0 + y1 * y1;
        blog[i] = __logf(b[i]);
    }
}

// ---------------------------------------------------------------- softmin step
// dir 0: f-side  (rows = x points, reduce over y)
// dir 1: g-side  (rows = y points, reduce over x)
// use_g: include dual/eps in h;  avg: output = 0.5*(old + new)
__global__ __launch_bounds__(128) void sk_softmin(
    const float* __restrict__ x,    const float* __restrict__ y,
    const float* __restrict__ sx,   const float* __restrict__ sy,
    const float* __restrict__ alog, const float* __restrict__ blog,
    const float* __restrict__ f_in, const float* __restrict__ g_in,
    float* __restrict__ f_out,      float* __restrict__ g_out,
    float eps, int use_g, int avg)
{
    __shared__ float  sh_h[NPTS];    // h'_m = h_m - 0.5*sy_m/eps
    __shared__ float2 sh_p[NPTS];    // src point * (1/eps)

    const int   bId   = blockIdx.x >> 4;    // batch
    const int   chunk = blockIdx.x & 15;    // 64-row chunk
    const int   dir   = blockIdx.y;
    const float inv_eps = 1.0f / eps;

    const float *dst_pts, *src_pts, *dst_sq, *src_sq, *hlog, *dual, *self_old;
    float* outp;
    if (dir == 0) { dst_pts = x; src_pts = y; dst_sq = sx; src_sq = sy;
                    hlog = blog; dual = g_in; self_old = f_in; outp = f_out; }
    else          { dst_pts = y; src_pts = x; dst_sq = sy; src_sq = sx;
                    hlog = alog; dual = f_in; self_old = g_in; outp = g_out; }

    const size_t base = (size_t)bId * NPTS;

    // stage the entire source side of this batch in LDS
    for (int m = threadIdx.x; m < NPTS; m += blockDim.x) {
        float p0 = src_pts[(base + m) * 2 + 0];
        float p1 = src_pts[(base + m) * 2 + 1];
        float h  = hlog[base + m];
        if (use_g) h += dual[base + m] * inv_eps;
        sh_p[m] = make_float2(p0 * inv_eps, p1 * inv_eps);
        sh_h[m] = h - 0.5f * inv_eps * src_sq[base + m];
    }
    __syncthreads();

    const int lane = threadIdx.x & 31;
    const int wave = threadIdx.x >> 5;
    const int n0   = chunk * 64 + wave * 16;     // this wave's 16 rows
    const int ncol = lane & 15;
    const int n    = n0 + ncol;

    // B operand (4x16, K x N): lanes 0-15 hold K=0,1 = (x0,x1); K=2,3 = 0
    v2f bop;
    if (lane < 16) { bop[0] = dst_pts[(base + n) * 2 + 0];
                     bop[1] = dst_pts[(base + n) * 2 + 1]; }
    else           { bop[0] = 0.0f; bop[1] = 0.0f; }

    const int hi8 = (lane >> 4) << 3;            // D-layout: M = r + 8*(lane>=16)
    float rmax = -INFINITY, rsum = 0.0f;

    for (int t = 0; t < NPTS / 16; ++t) {
        const int m0 = t << 4;

        // A operand (16x4, M x K): lanes 0-15 hold (y0,y1)/eps; K=2,3 = 0
        v2f aop;
        if (lane < 16) { float2 p = sh_p[m0 + lane]; aop[0] = p.x; aop[1] = p.y; }
        else           { aop[0] = 0.0f; aop[1] = 0.0f; }

        // C operand: broadcast h'_m per row m  (m = m0 + hi8 + r)
        v8f c;
        #pragma unroll
        for (int r = 0; r < 8; ++r) c[r] = sh_h[m0 + hi8 + r];

        // logits: t[m][n] = (x_n . y_m)/eps + h'_m
        v8f d = __builtin_amdgcn_wmma_f32_16x16x4_f32(
            false, aop, false, bop, (short)0, c, false, false);

        // online logsumexp over this lane's 8 m-values
        float tmax = d[0];
        #pragma unroll
        for (int r = 1; r < 8; ++r) tmax = fmaxf(tmax, d[r]);
        float nmax = fmaxf(rmax, tmax);
        float acc  = rsum * __expf(rmax - nmax);
        #pragma unroll
        for (int r = 0; r < 8; ++r) acc += __expf(d[r] - nmax);
        rsum = acc; rmax = nmax;
    }

    // merge the two lane halves (m offsets 0-7 vs 8-15 per tile)
    float omax = __shfl_xor(rmax, 16, 32);
    float osum = __shfl_xor(rsum, 16, 32);
    float m2   = fmaxf(rmax, omax);
    float s2   = rsum * __expf(rmax - m2) + osum * __expf(omax - m2);
    float lse  = m2 + __logf(s2);

    // f_t = -eps*(lse + r_n) with r_n = -0.5*|x_n|^2/eps  ->  0.5*|x_n|^2 - eps*lse
    float ft = 0.5f * dst_sq[base + n] - eps * lse;
    if (avg) ft = 0.5f * (self_old[base + n] + ft);
    if (lane < 16) outp[base + n] = ft;
}

// ---------------------------------------------------------------- final loss
__global__ __launch_bounds__(256) void sk_loss(
    const float* __restrict__ a, const float* __restrict__ bw,
    const float* __restrict__ f, const float* __restrict__ g,
    float* __restrict__ out)
{
    __shared__ float red[256];
    float acc = 0.0f;
    for (int i = threadIdx.x; i < BATCH * NPTS; i += 256)
        acc += a[i] * f[i] + bw[i] * g[i];
    red[threadIdx.x] = acc;
    __syncthreads();
    for (int s = 128; s > 0; s >>= 1) {
        if (threadIdx.x < s) red[threadIdx.x] += red[threadIdx.x + s];
        __syncthreads();
    }
    if (threadIdx.x == 0) out[0] = red[0] / (float)BATCH;
}

// ---------------------------------------------------------------- host
extern "C" void kernel_launch(void* const* d_in, const int* in_sizes, int n_in,
                              void* d_out, int out_size, void* d_ws, size_t ws_size,
                              hipStream_t stream) {
    const float* a = (const float*)d_in[0];   // [16,1024]
    const float* x = (const float*)d_in[1];   // [16,1024,2]
    const float* b = (const float*)d_in[2];   // [16,1024]
    const float* y = (const float*)d_in[3];   // [16,1024,2]
    float* out = (float*)d_out;

    float* ws   = (float*)d_ws;
    const int P = BATCH * NPTS;               // 16384
    float* sx   = ws + 0 * P;
    float* sy   = ws + 1 * P;
    float* alog = ws + 2 * P;
    float* blog = ws + 3 * P;
    float* fbuf[2] = { ws + 4 * P, ws + 5 * P };
    float* gbuf[2] = { ws + 6 * P, ws + 7 * P };

    // eps schedule: [1.0] + exp(arange(0, 2*ln(0.01), 2*ln(0.9))) + [1e-4]
    std::vector<float> eps;
    eps.push_back(1.0f);
    const double step = 2.0 * std::log(0.9), stop = 2.0 * std::log(0.01);
    for (int i = 0;; ++i) {
        double v = (double)i * step;
        if (!(v > stop)) break;
        eps.push_back((float)std::exp(v));
    }
    eps.push_back(1e-4f);

    sk_prep<<<P / 256, 256, 0, stream>>>(a, x, b, y, sx, sy, alog, blog);

    const dim3 grid(BATCH * (NPTS / 64), 2), block(128);

    // init: f = softmin(eps0, C, b_log); g symmetric   (h has no dual term)
    int cur = 0;
    sk_softmin<<<grid, block, 0, stream>>>(x, y, sx, sy, alog, blog,
        fbuf[1], gbuf[1], fbuf[cur], gbuf[cur], eps[0], /*use_g=*/0, /*avg=*/0);

    // annealed Jacobi iterations over the full schedule
    for (size_t i = 0; i < eps.size(); ++i) {
        int nxt = cur ^ 1;
        sk_softmin<<<grid, block, 0, stream>>>(x, y, sx, sy, alog, blog,
            fbuf[cur], gbuf[cur], fbuf[nxt], gbuf[nxt], eps[i], 1, /*avg=*/1);
        cur = nxt;
    }

    // final extrapolation at eps = blur^2 (no averaging)
    int fin = cur ^ 1;
    sk_softmin<<<grid, block, 0, stream>>>(x, y, sx, sy, alog, blog,
        fbuf[cur], gbuf[cur], fbuf[fin], gbuf[fin], eps.back(), 1, /*avg=*/0);

    sk_loss<<<1, 256, 0, stream>>>(a, b, fbuf[fin], gbuf[fin], out);
}